// SSMBlock_71906342470089
// MI455X (gfx1250) — compile-verified
//
#include <hip/hip_runtime.h>
#include <hip/hip_bf16.h>

typedef float v2f __attribute__((ext_vector_type(2)));
typedef float v8f __attribute__((ext_vector_type(8)));

#define TILE_M 128
#define TILE_N 128
#define TILE_K 16
#define LDT_S 20    // 16 + 4 pad (floats); 80B row stride keeps 16B alignment

// ---- gfx1250 async global->LDS copy (ASYNCcnt-tracked, bypasses VGPRs) ----
__device__ __forceinline__ void async_ld_b128(unsigned lds_byte_off, const float* gaddr) {
    asm volatile("global_load_async_to_lds_b128 %0, %1, off"
                 :: "v"(lds_byte_off), "v"(gaddr) : "memory");
}
__device__ __forceinline__ void wait_asynccnt0() {
    asm volatile("s_wait_asynccnt 0x0" ::: "memory");
}

// C[M,N] = A[M,K] @ W[N,K]^T + bias[N]   (all row-major, f32)
// Block: 256 threads (8 waves). Block tile 128x128, wave tile 32x64,
// 2x4 grid of 16x16 WMMA tiles per wave. A and W tiles are both staged to
// LDS in natural [row][k] layout via async b128 DMA (double buffered); the
// f32 16x16x4 A- and B-fragments then have identical ds_load_b64 patterns.
__global__ __launch_bounds__(256)
void wmma_gemm_bias(const float* __restrict__ A,
                    const float* __restrict__ W,
                    const float* __restrict__ bias,
                    float* __restrict__ C,
                    int M, int N, int K)
{
    __shared__ alignas(16) float sA[2][TILE_M * LDT_S];
    __shared__ alignas(16) float sB[2][TILE_M * LDT_S];

    const int tid  = threadIdx.x;
    const int lane = tid & 31;
    const int wave = tid >> 5;
    const int wm   = wave & 3;           // 4 waves along M
    const int wn   = wave >> 2;          // 2 waves along N
    const int m0   = blockIdx.y * TILE_M;
    const int n0   = blockIdx.x * TILE_N;

    // WMMA f32 16x16x4 fragment addressing (ISA 7.12.2):
    // A: lanes 0-15 hold M=lane, K={0,1};  lanes 16-31 hold M=lane-16, K={2,3}
    // B: lanes 0-15 hold N=lane, K={0,1};  lanes 16-31 hold N=lane-16, K={2,3}
    const int rsel = lane & 15;          // M (A-frag) / N (B-frag) within tile
    const int kb   = (lane >> 4) << 1;   // 0 or 2

    v8f acc[2][4];
    const v8f vzero = {0.f,0.f,0.f,0.f,0.f,0.f,0.f,0.f};
    #pragma unroll
    for (int i = 0; i < 2; ++i)
        #pragma unroll
        for (int j = 0; j < 4; ++j)
            acc[i][j] = vzero;

    // Staging: each thread DMAs 2x b128 of A and 2x b128 of W per stage.
    const int lrow = tid >> 2;           // 0..63
    const int lk4  = (tid & 3) << 2;     // 0,4,8,12

    const unsigned sA_base = (unsigned)(size_t)&sA[0][0];
    const unsigned sB_base = (unsigned)(size_t)&sB[0][0];

    auto stage = [&](int buf, int kt) {
        const int k0 = kt * TILE_K;
        const unsigned bufOff = (unsigned)(buf * (TILE_M * LDT_S)) * 4u;
        #pragma unroll
        for (int p = 0; p < 2; ++p) {
            const int row = lrow + p * 64;
            const unsigned toff = bufOff + (unsigned)(row * LDT_S + lk4) * 4u;
            async_ld_b128(sA_base + toff, A + (size_t)(m0 + row) * K + k0 + lk4);
            async_ld_b128(sB_base + toff, W + (size_t)(n0 + row) * K + k0 + lk4);
        }
    };

    const int kTiles = K / TILE_K;
    stage(0, 0);
    wait_asynccnt0();
    __syncthreads();

    for (int kt = 0; kt < kTiles; ++kt) {
        const int cur = kt & 1;
        if (kt + 1 < kTiles) stage(cur ^ 1, kt + 1);   // overlap DMA with WMMA
        if (kt + 2 < kTiles) {
            __builtin_prefetch(A + (size_t)(m0 + lrow) * K + (kt + 2) * TILE_K + lk4, 0, 1);
            __builtin_prefetch(W + (size_t)(n0 + lrow) * K + (kt + 2) * TILE_K + lk4, 0, 1);
        }

        const float* As = &sA[cur][0];
        const float* Bs = &sB[cur][0];
        #pragma unroll
        for (int ks = 0; ks < 4; ++ks) {
            const int k0 = (ks << 2) + kb;
            v2f a[2], b[4];
            #pragma unroll
            for (int i = 0; i < 2; ++i)
                a[i] = *(const v2f*)(As + (wm * 32 + i * 16 + rsel) * LDT_S + k0);
            #pragma unroll
            for (int j = 0; j < 4; ++j)
                b[j] = *(const v2f*)(Bs + (wn * 64 + j * 16 + rsel) * LDT_S + k0);
            #pragma unroll
            for (int i = 0; i < 2; ++i)
                #pragma unroll
                for (int j = 0; j < 4; ++j)
                    acc[i][j] = __builtin_amdgcn_wmma_f32_16x16x4_f32(
                        false, a[i], false, b[j], (short)0, acc[i][j], false, false);
        }
        if (kt + 1 < kTiles) wait_asynccnt0();         // next buffer resident
        __syncthreads();
    }

    // C/D layout: VGPR r -> M=r (lanes 0-15), M=r+8 (lanes 16-31); N = lane&15
    const int half8 = (lane >> 4) << 3;
    #pragma unroll
    for (int j = 0; j < 4; ++j) {
        const int gc = n0 + wn * 64 + j * 16 + rsel;
        const float bv = bias[gc];
        #pragma unroll
        for (int i = 0; i < 2; ++i) {
            #pragma unroll
            for (int r = 0; r < 8; ++r) {
                const int gr = m0 + wm * 32 + i * 16 + half8 + r;
                C[(size_t)gr * N + gc] = acc[i][j][r] + bv;
            }
        }
    }
}

// Fused depthwise-causal-conv(k=4) + SiLU + SSM scan + SiLU gate.
// One thread per (b, d): 4-tap window and 16 SSM states live in registers,
// lanes cover consecutive d -> coalesced column walks over t.
__global__ __launch_bounds__(256)
void conv_scan_gate(const float* __restrict__ xp,     // [B*L, 2*DI]: x1 | x2
                    const float* __restrict__ conv_w, // [DI, 4]
                    const float* __restrict__ conv_b, // [DI]
                    const float* __restrict__ A_log,  // [DI, 16]
                    const float* __restrict__ Bp,     // [DI, 16]
                    const float* __restrict__ Cp,     // [DI, 16]
                    const float* __restrict__ log_dt, // [DI]
                    float* __restrict__ yg,           // [B*L, DI] (gated)
                    int Bn, int L, int DI)
{
    const int idx = blockIdx.x * blockDim.x + threadIdx.x;
    if (idx >= Bn * DI) return;
    const int d = idx % DI;
    const int b = idx / DI;

    const float dt = __expf(log_dt[d]);
    const float w0 = conv_w[d * 4 + 0];
    const float w1 = conv_w[d * 4 + 1];
    const float w2 = conv_w[d * 4 + 2];
    const float w3 = conv_w[d * 4 + 3];
    const float cb = conv_b[d];

    float dA[16], dB[16], Cc[16], h[16];
    #pragma unroll
    for (int n = 0; n < 16; ++n) {
        const float a = __expf(A_log[d * 16 + n]);   // A = exp(A_log)
        dA[n] = __expf(dt * a);                      // dA = exp(dt*A)
        dB[n] = dt * Bp[d * 16 + n];                 // dB = dt*B
        Cc[n] = Cp[d * 16 + n];
        h[n]  = 0.f;
    }

    const size_t strideX = (size_t)(2 * DI);
    const float* x1c = xp + (size_t)b * L * strideX + d;
    const float* x2c = x1c + DI;
    float*       yc  = yg + (size_t)b * L * DI + d;

    float xm3 = 0.f, xm2 = 0.f, xm1 = 0.f;          // causal zero left-pad
    for (int t = 0; t < L; ++t) {
        const float x0 = x1c[(size_t)t * strideX];
        float c = fmaf(w3, x0, fmaf(w2, xm1, fmaf(w1, xm2, fmaf(w0, xm3, cb))));
        xm3 = xm2; xm2 = xm1; xm1 = x0;
        const float ut = c / (1.f + __expf(-c));     // SiLU

        float y = 0.f;
        #pragma unroll
        for (int n = 0; n < 16; ++n) {
            h[n] = fmaf(dA[n], h[n], dB[n] * ut);
            y    = fmaf(h[n], Cc[n], y);
        }
        const float g = x2c[(size_t)t * strideX];
        yc[(size_t)t * DI] = y * (g / (1.f + __expf(-g)));  // gate = SiLU(x2)
    }
}

extern "C" void kernel_launch(void* const* d_in, const int* in_sizes, int n_in,
                              void* d_out, int out_size, void* d_ws, size_t ws_size,
                              hipStream_t stream) {
    (void)in_sizes; (void)n_in; (void)out_size; (void)ws_size;
    const float* x    = (const float*)d_in[0];
    const float* Win  = (const float*)d_in[1];   // [4096, 1024]
    const float* bin  = (const float*)d_in[2];   // [4096]
    const float* cw   = (const float*)d_in[3];   // [2048,1,4]
    const float* cb   = (const float*)d_in[4];   // [2048]
    const float* Alg  = (const float*)d_in[5];   // [2048,16]
    const float* Bp   = (const float*)d_in[6];   // [2048,16]
    const float* Cp   = (const float*)d_in[7];   // [2048,16]
    const float* ldt  = (const float*)d_in[8];   // [2048]
    const float* Wout = (const float*)d_in[9];   // [1024, 2048]
    const float* bout = (const float*)d_in[10];  // [1024]
    float* out = (float*)d_out;

    const int Bb = 4, L = 2048, Dm = 1024, DI = 2048;
    const int M = Bb * L;                        // 8192 rows

    float* xp = (float*)d_ws;                    // [M, 4096]
    float* yg = xp + (size_t)M * (2 * DI);       // [M, 2048]

    dim3 blk(256);
    // xp = x @ Win^T + bin
    wmma_gemm_bias<<<dim3((2 * DI) / TILE_N, M / TILE_M), blk, 0, stream>>>(
        x, Win, bin, xp, M, 2 * DI, Dm);
    // yg = silu(x2) * scan(silu(conv(x1)))
    conv_scan_gate<<<(Bb * DI + 255) / 256, blk, 0, stream>>>(
        xp, cw, cb, Alg, Bp, Cp, ldt, yg, Bb, L, DI);
    // out = yg @ Wout^T + bout
    wmma_gemm_bias<<<dim3(Dm / TILE_N, M / TILE_M), blk, 0, stream>>>(
        yg, Wout, bout, out, M, Dm, DI);
}